// EdgeNetwork_36490042146902
// MI455X (gfx1250) — compile-verified
//
#include <hip/hip_runtime.h>
#include <math.h>

typedef __attribute__((ext_vector_type(2))) float v2f;
typedef __attribute__((ext_vector_type(4))) float v4f;
typedef __attribute__((ext_vector_type(8))) float v8f;

#define DHID   128
#define TILE_M 16
#define WPB    6            // waves per block (wave32) -> 192 threads
#define HSTR   132          // h-tile row stride in floats (132 % 64 == 4 -> conflict-free)
#define LNEPS  1e-5f

#if __has_builtin(__builtin_amdgcn_tanhf)
#define TANHF(v) __builtin_amdgcn_tanhf(v)   // v_tanh_f32 (single TRANS op)
#else
#define TANHF(v) tanhf(v)
#endif

// LDS float offsets: packed W0 (32768) | W1 (16384) | W2 (16384) | 6 h-buffers
#define OFF_W1 32768
#define OFF_W2 49152
#define OFF_H  65536
#define SMEM_FLOATS (OFF_H + WPB * TILE_M * HSTR)   // 78208 floats = 305.5 KB

// ---------------------------------------------------------------------------
// Pack row-major W[K x 128] into WMMA-B-fragment-native LDS layout:
//   dst[kk*512 + nn*64 + c*4 + sub],  kk = k/4, sub = k%4 (= 2g+i), col = c+16nn
// Reader (lane g,c) then gets {W[k0][col], W[k0+1][col]} as ONE conflict-free
// ds_load_b64: group0 occupies banks {4c,4c+1}, group1 {4c+2,4c+3}.
// ---------------------------------------------------------------------------
__device__ __forceinline__ void pack_w(float* __restrict__ dst,
                                       const float* __restrict__ W,
                                       int K, int tid, int nthr)
{
    const int total = K << 7;                 // K*128
    for (int i = tid; i < total; i += nthr) {
        const int r   = i >> 7;               // row k
        const int col = i & 127;
        dst[((r >> 2) << 9) + ((col >> 4) << 6) + ((col & 15) << 2) + (r & 3)] = W[i];
    }
}

// ---------------------------------------------------------------------------
// [16 x 4*KSTEPS] (A from LDS h-buffer, stride HS) x [4*KSTEPS x 128] (packed B
// in LDS) -> 8 v8f accumulators, via V_WMMA_F32_16X16X4_F32.
// ---------------------------------------------------------------------------
template <int KSTEPS, int HS>
__device__ __forceinline__ void gemm_lds(const float* __restrict__ hbuf,
                                         const float* __restrict__ wp,
                                         v8f acc[8], int lane)
{
    const int row = lane & 15;
    const int g2  = (lane >> 4) << 1;         // 0 or 2
    const float* wl = wp + ((lane & 15) << 2) + g2;
#pragma unroll 4
    for (int kk = 0; kk < KSTEPS; ++kk) {
        const v2f a = *(const v2f*)(hbuf + row * HS + (kk << 2) + g2); // ds_load_b64
        const float* wk = wl + (kk << 9);
#pragma unroll
        for (int nn = 0; nn < 8; ++nn) {
            const v2f b = *(const v2f*)(wk + (nn << 6));               // ds_load_b64
            acc[nn] = __builtin_amdgcn_wmma_f32_16x16x4_f32(
                false, a, false, b, (short)0, acc[nn], false, false);
        }
    }
}

// bias + LayerNorm + tanh on the 16x128 register tile (C/D layout)
__device__ __forceinline__ void layernorm_tanh(v8f acc[8],
                                               const float* __restrict__ bias,
                                               const float* __restrict__ gamma,
                                               const float* __restrict__ beta,
                                               int lane)
{
    const int col0 = lane & 15;
    float bsv[8], gv[8], bv[8];
#pragma unroll
    for (int nn = 0; nn < 8; ++nn) {
        const int c = col0 + (nn << 4);
        bsv[nn] = bias[c]; gv[nn] = gamma[c]; bv[nn] = beta[c];
    }
#pragma unroll
    for (int j = 0; j < 8; ++j) {
        float s = 0.f, ss = 0.f;
#pragma unroll
        for (int nn = 0; nn < 8; ++nn) {
            float v = acc[nn][j] + bsv[nn];
            acc[nn][j] = v;
            s += v;
            ss = fmaf(v, v, ss);
        }
#pragma unroll
        for (int m = 1; m < 16; m <<= 1) {     // xor<16: stays in 16-lane groups
            s  += __shfl_xor(s,  m, 32);
            ss += __shfl_xor(ss, m, 32);
        }
        const float mu  = s * (1.0f / 128.0f);
        const float var = fmaf(-mu, mu, ss * (1.0f / 128.0f));
        const float inv = rsqrtf(var + LNEPS);
#pragma unroll
        for (int nn = 0; nn < 8; ++nn) {
            const float v = (acc[nn][j] - mu) * inv;
            acc[nn][j] = TANHF(fmaf(v, gv[nn], bv[nn]));
        }
    }
}

// store 16x128 register tile to LDS h-buffer (stride HSTR, conflict-free)
__device__ __forceinline__ void store_tile(float* __restrict__ hbuf,
                                           const v8f acc[8], int lane)
{
    const int col0 = lane & 15;
    const int grp  = lane >> 4;
#pragma unroll
    for (int j = 0; j < 8; ++j) {
        const int row = j + (grp << 3);
#pragma unroll
        for (int nn = 0; nn < 8; ++nn)
            hbuf[row * HSTR + col0 + (nn << 4)] = acc[nn][j];
    }
}

// gather 16 node rows (128 f32) into the h-buffer, coalesced b128
__device__ __forceinline__ void stage_rows(float* __restrict__ hbuf,
                                           const float* __restrict__ x,
                                           const int* __restrict__ idxbase,
                                           int e0, int E, int lane)
{
    const int c4 = lane << 2;
#pragma unroll 4
    for (int r = 0; r < TILE_M; ++r) {
        int e = e0 + r; if (e >= E) e = E - 1;           // uniform clamp
        const int n = idxbase[e];
        const v4f v = *(const v4f*)(x + (size_t)n * DHID + c4);
        *(v4f*)(hbuf + r * HSTR + c4) = v;
    }
}

__global__ void __launch_bounds__(WPB * 32, 1)
edge_mlp_wmma_f32(const float* __restrict__ x, const int* __restrict__ eidx, int E,
                  const float* __restrict__ W0, const float* __restrict__ b0,
                  const float* __restrict__ W1, const float* __restrict__ b1,
                  const float* __restrict__ W2, const float* __restrict__ b2,
                  const float* __restrict__ W3, const float* __restrict__ b3,
                  const float* __restrict__ g0, const float* __restrict__ be0,
                  const float* __restrict__ g1, const float* __restrict__ be1,
                  const float* __restrict__ g2, const float* __restrict__ be2,
                  float* __restrict__ out)
{
    extern __shared__ float smem[];
    const int tid  = threadIdx.x;
    const int lane = tid & 31;
    const int wave = tid >> 5;

    float* wp0  = smem;
    float* wp1  = smem + OFF_W1;
    float* wp2  = smem + OFF_W2;
    float* hbuf = smem + OFF_H + wave * (TILE_M * HSTR);

    // ---- stage packed weights once per block ----
    pack_w(wp0, W0, 2 * DHID, tid, WPB * 32);
    pack_w(wp1, W1, DHID,     tid, WPB * 32);
    pack_w(wp2, W2, DHID,     tid, WPB * 32);
    __syncthreads();

    const int tiles  = (E + TILE_M - 1) / TILE_M;
    const int stride = gridDim.x * WPB;
    const int sweeps = (tiles + stride - 1) / stride;    // uniform trip count

    for (int it = 0; it < sweeps; ++it) {
        const int tile = (it * gridDim.x + blockIdx.x) * WPB + wave;
        const int e0   = tile * TILE_M;

        // prefetch next sweep's edge-index lines (global_prefetch_b8)
        {
            const int tn = tile + stride;
            if (tn < tiles) {
                __builtin_prefetch(&eidx[tn * TILE_M], 0, 1);
                __builtin_prefetch(&eidx[E + tn * TILE_M], 0, 1);
            }
        }

        // ---- layer 0, split-K: x[start] half then x[end] half ----
        v8f h0[8] = {};
        stage_rows(hbuf, x, eidx, e0, E, lane);          // x[start] rows
        __syncthreads();
        gemm_lds<32, HSTR>(hbuf, wp0, h0, lane);         // k = 0..127
        __syncthreads();
        stage_rows(hbuf, x, eidx + E, e0, E, lane);      // x[end] rows
        __syncthreads();
        gemm_lds<32, HSTR>(hbuf, wp0 + 32 * 512, h0, lane); // k = 128..255
        layernorm_tanh(h0, b0, g0, be0, lane);
        store_tile(hbuf, h0, lane);
        __syncthreads();

        // ---- layer 1 ----
        v8f h1[8] = {};
        gemm_lds<32, HSTR>(hbuf, wp1, h1, lane);
        layernorm_tanh(h1, b1, g1, be1, lane);
        store_tile(hbuf, h1, lane);
        __syncthreads();

        // ---- layer 2 ----
        v8f h2[8] = {};
        gemm_lds<32, HSTR>(hbuf, wp2, h2, lane);
        layernorm_tanh(h2, b2, g2, be2, lane);

        // ---- layer 3: 128 -> 1 from registers ----
        {
            const int col0 = lane & 15;
            const int grp  = lane >> 4;
            float w3[8];
#pragma unroll
            for (int nn = 0; nn < 8; ++nn) w3[nn] = W3[col0 + (nn << 4)];
            const float bias3 = b3[0];
#pragma unroll
            for (int j = 0; j < 8; ++j) {
                float p = 0.f;
#pragma unroll
                for (int nn = 0; nn < 8; ++nn) p = fmaf(h2[nn][j], w3[nn], p);
#pragma unroll
                for (int m = 1; m < 16; m <<= 1) p += __shfl_xor(p, m, 32);
                const int e = e0 + j + (grp << 3);
                if (col0 == j && e < E) out[e] = p + bias3;
            }
        }
        __syncthreads();   // protect h-buffer before next sweep's staging
    }
}

extern "C" void kernel_launch(void* const* d_in, const int* in_sizes, int n_in,
                              void* d_out, int out_size, void* d_ws, size_t ws_size,
                              hipStream_t stream)
{
    const float* x   = (const float*)d_in[0];
    const int*   ei  = (const int*)  d_in[1];
    const float* W0  = (const float*)d_in[2];
    const float* b0  = (const float*)d_in[3];
    const float* W1  = (const float*)d_in[4];
    const float* b1  = (const float*)d_in[5];
    const float* W2  = (const float*)d_in[6];
    const float* b2  = (const float*)d_in[7];
    const float* W3  = (const float*)d_in[8];
    const float* b3  = (const float*)d_in[9];
    const float* g0  = (const float*)d_in[10];
    const float* be0 = (const float*)d_in[11];
    const float* g1  = (const float*)d_in[12];
    const float* be1 = (const float*)d_in[13];
    const float* g2  = (const float*)d_in[14];
    const float* be2 = (const float*)d_in[15];

    const int E     = in_sizes[1] / 2;
    const int tiles = (E + TILE_M - 1) / TILE_M;
    int blocks = (tiles + WPB - 1) / WPB;
    if (blocks > 1024) blocks = 1024;          // grid-stride; amortize weight staging
    const size_t shmem = (size_t)SMEM_FLOATS * sizeof(float);  // ~305.5 KB

    edge_mlp_wmma_f32<<<blocks, WPB * 32, shmem, stream>>>(
        x, ei, E, W0, b0, W1, b1, W2, b2, W3, b3,
        g0, be0, g1, be1, g2, be2, (float*)d_out);
}